// CantorGlobalAttention_33002528702702
// MI455X (gfx1250) — compile-verified
//
#include <hip/hip_runtime.h>
#include <math.h>

typedef float v2f __attribute__((ext_vector_type(2)));
typedef float v4f __attribute__((ext_vector_type(4)));
typedef float v8f __attribute__((ext_vector_type(8)));

#define ND 5      // directions
#define NE 16     // experts
#define NB 64     // batch
#define NP 4096   // patches per expert
#define NW 3      // local window

#define LOG2E 1.4426950408889634f

// ---------------------------------------------------------------------------
// Phase 1: row reductions Ksum[d,e,b] = sum_p K, Vmean[d,e,b] = mean_p V
// via V_WMMA_F32_16X16X4_F32 with B = ones (fp32 accumulate, RNE).
// One b128 load per lane feeds two WMMAs: since a row-sum is permutation
// invariant, WMMA step A sums window {p0,p1,p4,p5} (lanes 0-15 give k0,k1;
// lanes 16-31 give k2,k3) and step B sums {p2,p3,p6,p7}; a lane's data for
// both steps is its contiguous float4.
// Block = 256 threads = 8 waves; each block owns one (d, e, 16-row b-tile)
// of either K or V; each wave reduces a 512-wide p-slice; waves combine in LDS.
// ---------------------------------------------------------------------------
__global__ __launch_bounds__(256)
void cga_reduce_kernel(const float* __restrict__ K,
                       const float* __restrict__ V,
                       float* __restrict__ ksum,
                       float* __restrict__ vmean) {
    __shared__ float red[8][16];

    const int tile  = blockIdx.x;
    const int isV   = tile & 1;
    const int t     = tile >> 1;
    const int btile = t & 3;           // B/16 = 4 tiles
    const int e     = (t >> 2) & 15;
    const int d     = t >> 6;

    const float* __restrict__ src = isV ? V : K;

    const int lane  = threadIdx.x & 31;
    const int wave  = threadIdx.x >> 5;
    const int m     = lane & 15;       // A-matrix row (b within tile)
    const int khalf = lane >> 4;       // lanes 16-31 hold k=2,3

    const float* row = src + (size_t)((d * NE + e) * NB + btile * 16 + m) * NP;
    // each wave owns 512 p; lane reads float4 at (8-wide window) + khalf*4
    const float* lp = row + wave * (NP / 8) + khalf * 4;

    const v2f ones = {1.0f, 1.0f};
    v8f acc0 = {}, acc1 = {}, acc2 = {}, acc3 = {};

    // 64 8-p windows per wave, two windows (one b128 load each) per iteration.
    for (int it = 0; it < 32; ++it) {
        v4f f0 = *(const v4f*)(lp + it * 16);
        v4f f1 = *(const v4f*)(lp + it * 16 + 8);
        v2f a;
        a.x = f0.x; a.y = f0.y;
        acc0 = __builtin_amdgcn_wmma_f32_16x16x4_f32(
            false, a, false, ones, (short)0, acc0, false, false);
        a.x = f0.z; a.y = f0.w;
        acc1 = __builtin_amdgcn_wmma_f32_16x16x4_f32(
            false, a, false, ones, (short)0, acc1, false, false);
        a.x = f1.x; a.y = f1.y;
        acc2 = __builtin_amdgcn_wmma_f32_16x16x4_f32(
            false, a, false, ones, (short)0, acc2, false, false);
        a.x = f1.z; a.y = f1.w;
        acc3 = __builtin_amdgcn_wmma_f32_16x16x4_f32(
            false, a, false, ones, (short)0, acc3, false, false);
    }

    // D layout: lanes 0-15 VGPR r -> M=r ; lanes 16-31 VGPR r -> M=8+r.
    // Columns are replicated, so lane 0 / lane 16 carry the row sums.
    if (lane == 0 || lane == 16) {
        const int mo = khalf * 8;
#pragma unroll
        for (int r = 0; r < 8; ++r)
            red[wave][mo + r] = (acc0[r] + acc1[r]) + (acc2[r] + acc3[r]);
    }
    __syncthreads();

    if (threadIdx.x < 16) {
        float s = 0.0f;
#pragma unroll
        for (int wv = 0; wv < 8; ++wv) s += red[wv][threadIdx.x];
        const int b   = btile * 16 + threadIdx.x;
        const int idx = (d * NE + e) * NB + b;
        if (isV) vmean[idx] = s * (1.0f / (float)NP);
        else     ksum[idx]  = s;
    }
}

// ---------------------------------------------------------------------------
// Phase 1.5: fold routing + beta gating + scale + log2(e) into coefficients
//   Cc[d,e,b,w] = Ksum[d, routes[e,w], b] * scale * beta(e,w) * log2(e)
//   Vv[d,e,b,w] = Vmean[d, routes[e,w], b]
// (base-2 softmax == base-e softmax; lets phase 2 use raw v_exp_f32)
// ---------------------------------------------------------------------------
__global__ void cga_coef_kernel(const float* __restrict__ ksum,
                                const float* __restrict__ vmean,
                                const float* __restrict__ betas,
                                const float* __restrict__ temperature,
                                const int*   __restrict__ routes,
                                float* __restrict__ Cc,
                                float* __restrict__ Vv) {
    const int idx = blockIdx.x * blockDim.x + threadIdx.x;
    if (idx >= ND * NE * NB) return;
    const int b = idx & 63;
    const int e = (idx >> 6) & 15;
    const int d = idx >> 10;

    const float scale = 1.0f / (sqrtf(128.0f) * fabsf(temperature[0]));
#pragma unroll
    for (int w = 0; w < NW; ++w) {
        const int r = routes[e * NW + w];
        const float beta = (r == e)
            ? 1.0f
            : 1.0f / (1.0f + expf(-betas[e * NE + r]));
        const int nidx = (d * NE + r) * NB + b;
        Cc[idx * NW + w] = ksum[nidx] * scale * beta * LOG2E;
        Vv[idx * NW + w] = vmean[nidx];
    }
}

// ---------------------------------------------------------------------------
// Phase 2: stream Q once (b128 per thread); 3-way stable base-2 softmax and
// the direction mean in-register (no atomics -> deterministic).
// Block -> (e, b, 1024-wide p-tile); coefficient loads are workgroup-uniform.
// ---------------------------------------------------------------------------
__global__ __launch_bounds__(256)
void cga_main_kernel(const float* __restrict__ Q,
                     const float* __restrict__ Cc,
                     const float* __restrict__ Vv,
                     float* __restrict__ out) {
    const int ptile = blockIdx.x & 3;
    const int b     = (blockIdx.x >> 2) & 63;
    const int e     = blockIdx.x >> 8;
    const int p     = ptile * 1024 + threadIdx.x * 4;

    v4f acc = {};
#pragma unroll
    for (int d = 0; d < ND; ++d) {
        const int cidx = ((d * NE + e) * NB + b) * NW;
        const float c0 = Cc[cidx + 0], c1 = Cc[cidx + 1], c2 = Cc[cidx + 2];
        const float v0 = Vv[cidx + 0], v1 = Vv[cidx + 1], v2 = Vv[cidx + 2];

        const v4f q = *(const v4f*)(Q + (size_t)((d * NE + e) * NB + b) * NP + p);
#pragma unroll
        for (int j = 0; j < 4; ++j) {
            const float s0 = q[j] * c0, s1 = q[j] * c1, s2 = q[j] * c2;
            const float mx = fmaxf(s0, fmaxf(s1, s2));
            const float e0 = __builtin_amdgcn_exp2f(s0 - mx);
            const float e1 = __builtin_amdgcn_exp2f(s1 - mx);
            const float e2 = __builtin_amdgcn_exp2f(s2 - mx);
            acc[j] += (e0 * v0 + e1 * v1 + e2 * v2) / (e0 + e1 + e2);
        }
    }
    v4f r;
#pragma unroll
    for (int j = 0; j < 4; ++j) r[j] = acc[j] * 0.2f;
    *(v4f*)(out + (size_t)b * (NE * NP) + (size_t)e * NP + p) = r;
}

// ---------------------------------------------------------------------------
extern "C" void kernel_launch(void* const* d_in, const int* in_sizes, int n_in,
                              void* d_out, int out_size, void* d_ws, size_t ws_size,
                              hipStream_t stream) {
    (void)in_sizes; (void)n_in; (void)out_size; (void)ws_size;

    const float* Q      = (const float*)d_in[0];
    const float* K      = (const float*)d_in[1];
    const float* V      = (const float*)d_in[2];
    const float* betas  = (const float*)d_in[3];
    const float* temp   = (const float*)d_in[4];
    const int*   routes = (const int*)d_in[5];
    float* out = (float*)d_out;

    float* ws    = (float*)d_ws;
    float* ksum  = ws;                       // 5120 floats
    float* vmean = ws + ND * NE * NB;        // 5120 floats
    float* Cc    = ws + 2 * ND * NE * NB;    // 15360 floats
    float* Vv    = Cc + ND * NE * NB * NW;   // 15360 floats

    // Phase 1: 5*16*4 b-tiles * 2 tensors = 640 blocks of 8 waves.
    cga_reduce_kernel<<<ND * NE * (NB / 16) * 2, 256, 0, stream>>>(K, V, ksum, vmean);

    // Phase 1.5: 5120 coefficient sets.
    cga_coef_kernel<<<(ND * NE * NB + 255) / 256, 256, 0, stream>>>(
        ksum, vmean, betas, temp, routes, Cc, Vv);

    // Phase 2: 4 p-tiles * 64 b * 16 e = 4096 blocks.
    cga_main_kernel<<<NE * NB * (NP / 1024), 256, 0, stream>>>(Q, Cc, Vv, out);
}